// EdgeGAT_60421599920310
// MI455X (gfx1250) — compile-verified
//
#include <hip/hip_runtime.h>
#include <hip/hip_bf16.h>
#include <math.h>

// ---------------- problem constants (match reference) ----------------
#define N_NODES   20000
#define N_EDGES   320000
#define IN_DIM    512
#define HEADS     4
#define CHAN      128
#define HC        512            // HEADS*CHAN
#define EDIM      4
#define TOT_E     (N_EDGES + N_NODES)   // edges + self loops = 340000
#define NEG_SLOPE 0.2f
#define BN_EPS    1e-5f

typedef __attribute__((ext_vector_type(2))) float v2f;
typedef __attribute__((ext_vector_type(8))) float v8f;

// ---------------- helpers ----------------
__device__ __forceinline__ void atomicMaxF(float* addr, float val) {
  // standard monotonic-bit-pattern trick; addr initialized to -inf
  if (val >= 0.0f) atomicMax((int*)addr, __float_as_int(val));
  else             atomicMin((unsigned int*)addr, __float_as_uint(val));
}

// ---------------- 0) init all accumulators ----------------
__global__ void init_kernel(float* __restrict__ out, float* __restrict__ amax,
                            float* __restrict__ denom, float* __restrict__ chsum,
                            float* __restrict__ chsq, float* __restrict__ easum) {
  const size_t i0 = (size_t)blockIdx.x * blockDim.x + threadIdx.x;
  const size_t stride = (size_t)gridDim.x * blockDim.x;
  for (size_t i = i0; i < (size_t)N_NODES * HC; i += stride) out[i] = 0.0f;
  for (size_t i = i0; i < (size_t)N_NODES * HEADS; i += stride) {
    amax[i] = -INFINITY; denom[i] = 0.0f;
  }
  for (size_t i = i0; i < HC; i += stride) { chsum[i] = 0.0f; chsq[i] = 0.0f; }
  if (i0 < EDIM) easum[i0] = 0.0f;
}

// ---------------- 1) h = x @ W via V_WMMA_F32_16X16X4_F32 ----------------
// One wave computes a 16x128 strip of h: 8 fp32 accumulators, K loop of 128
// wmma ops each. Fragment layout per CDNA5 ISA 7.12.2:
//   A (16x4, MxK): lane = m (0..15) | m (16..31); VGPR0/1 = K{k0+2*half, +1}
//   B (4x16, KxN): lane = n; same K split across lane halves
//   C/D (16x16):   VGPR v -> row v (lanes 0-15) / v+8 (lanes 16-31), col = lane&15
__global__ void __launch_bounds__(256) gemm_h_kernel(const float* __restrict__ x,
                                                     const float* __restrict__ W,
                                                     float* __restrict__ h) {
  const int wave = blockIdx.x * (blockDim.x >> 5) + (threadIdx.x >> 5);
  const int lane = threadIdx.x & 31;
  const int rt = wave >> 2;            // 16-row tile index, 0..1249 (20000/16)
  const int cg = wave & 3;             // 128-col group, 0..3
  if (rt >= N_NODES / 16) return;

  const int m    = lane & 15;
  const int half = lane >> 4;          // 0: K{0,1}, 1: K{2,3}
  const int row  = rt * 16 + m;
  const int colBase = cg * 128;

  v8f acc[8] = {};
  const float* xrow = x + (size_t)row * IN_DIM;

  for (int k0 = 0; k0 < IN_DIM; k0 += 4) {
    const int ka = k0 + half * 2;
    v2f a;
    a.x = xrow[ka];
    a.y = xrow[ka + 1];
    const float* Wr0 = W + (size_t)ka * HC;
    const float* Wr1 = Wr0 + HC;
#pragma unroll
    for (int t = 0; t < 8; ++t) {
      const int col = colBase + t * 16 + m;
      v2f b;
      b.x = Wr0[col];
      b.y = Wr1[col];
      acc[t] = __builtin_amdgcn_wmma_f32_16x16x4_f32(
          /*neg_a=*/false, a, /*neg_b=*/false, b,
          /*c_mod=*/(short)0, acc[t], /*reuse_a=*/false, /*reuse_b=*/false);
    }
  }
#pragma unroll
  for (int t = 0; t < 8; ++t) {
#pragma unroll
    for (int v = 0; v < 8; ++v) {
      const int r = rt * 16 + v + half * 8;
      const int c = colBase + t * 16 + m;
      h[(size_t)r * HC + c] = acc[t][v];
    }
  }
}

// ---------------- 2) sum of edge_attr over E (for self-loop fill='mean') ----
__global__ void easum_kernel(const float* __restrict__ ea, float* __restrict__ easum) {
  __shared__ float sm[256][EDIM];
  float loc[EDIM] = {0.f, 0.f, 0.f, 0.f};
  const int stride = gridDim.x * blockDim.x;
  for (int e = blockIdx.x * blockDim.x + threadIdx.x; e < N_EDGES; e += stride) {
#pragma unroll
    for (int d = 0; d < EDIM; ++d) loc[d] += ea[(size_t)e * EDIM + d];
  }
#pragma unroll
  for (int d = 0; d < EDIM; ++d) sm[threadIdx.x][d] = loc[d];
  __syncthreads();
  for (int off = 128; off > 0; off >>= 1) {
    if ((int)threadIdx.x < off)
#pragma unroll
      for (int d = 0; d < EDIM; ++d) sm[threadIdx.x][d] += sm[threadIdx.x + off][d];
    __syncthreads();
  }
  if (threadIdx.x == 0)
#pragma unroll
    for (int d = 0; d < EDIM; ++d) atomicAdd(&easum[d], sm[0][d]);
}

// ---------------- 3) tiny setup: M[d][h] and self-loop a_edge ----------------
// a_edge = (ea@W_edge).reshape(-1,H,C) . att_edge  ==  ea @ M,
// M[d,h] = sum_c W_edge[d, h*C+c] * att_edge[h,c]   (4x4 matrix)
__global__ void setup_small_kernel(const float* __restrict__ W_edge,
                                   const float* __restrict__ att_edge,
                                   const float* __restrict__ easum,
                                   float* __restrict__ M,
                                   float* __restrict__ loop_ae) {
  const int t = threadIdx.x;
  if (t < EDIM * HEADS) {
    const int d = t / HEADS, hd = t % HEADS;
    float s = 0.0f;
    for (int c = 0; c < CHAN; ++c)
      s += W_edge[(size_t)d * HC + hd * CHAN + c] * att_edge[hd * CHAN + c];
    M[d * HEADS + hd] = s;
  }
  __syncthreads();
  if (t < HEADS) {
    float s = 0.0f;
    for (int d = 0; d < EDIM; ++d)
      s += (easum[d] / (float)N_EDGES) * M[d * HEADS + t];
    loop_ae[t] = s;
  }
}

// ---------------- 4) a_src / a_dst: one wave per (node, head) ----------------
__global__ void __launch_bounds__(256) srcdst_kernel(const float* __restrict__ h,
    const float* __restrict__ att_src, const float* __restrict__ att_dst,
    float* __restrict__ a_src, float* __restrict__ a_dst) {
  const int wave = blockIdx.x * 8 + (threadIdx.x >> 5);
  const int lane = threadIdx.x & 31;
  if (wave >= N_NODES * HEADS) return;
  const int n = wave / HEADS, hd = wave % HEADS;
  const float* hp = h + (size_t)n * HC + hd * CHAN;
  const float* as = att_src + hd * CHAN;
  const float* ad = att_dst + hd * CHAN;
  float ss = 0.0f, sd = 0.0f;
#pragma unroll
  for (int i = 0; i < CHAN; i += 32) {
    const float hv = hp[i + lane];
    ss += hv * as[i + lane];
    sd += hv * ad[i + lane];
  }
#pragma unroll
  for (int off = 16; off > 0; off >>= 1) {
    ss += __shfl_down(ss, off, 32);
    sd += __shfl_down(sd, off, 32);
  }
  if (lane == 0) { a_src[wave] = ss; a_dst[wave] = sd; }
}

// ---------------- 5) alpha = leaky(a_src[s]+a_dst[d]+a_edge); segment max ----
__global__ void alpha_max_kernel(const long long* __restrict__ ei,
    const float* __restrict__ ea, const float* __restrict__ M,
    const float* __restrict__ loop_ae,
    const float* __restrict__ a_src, const float* __restrict__ a_dst,
    float* __restrict__ alpha, float* __restrict__ amax) {
  const int e = blockIdx.x * blockDim.x + threadIdx.x;
  if (e >= TOT_E) return;
  long long s, d;
  float ae[HEADS];
  if (e < N_EDGES) {
    s = ei[e]; d = ei[N_EDGES + e];
    float v[EDIM];
#pragma unroll
    for (int dd = 0; dd < EDIM; ++dd) v[dd] = ea[(size_t)e * EDIM + dd];
#pragma unroll
    for (int hd = 0; hd < HEADS; ++hd) {
      float acc = 0.0f;
#pragma unroll
      for (int dd = 0; dd < EDIM; ++dd) acc += v[dd] * M[dd * HEADS + hd];
      ae[hd] = acc;
    }
  } else {
    s = d = (long long)(e - N_EDGES);
#pragma unroll
    for (int hd = 0; hd < HEADS; ++hd) ae[hd] = loop_ae[hd];
  }
#pragma unroll
  for (int hd = 0; hd < HEADS; ++hd) {
    float al = a_src[s * HEADS + hd] + a_dst[d * HEADS + hd] + ae[hd];
    al = (al > 0.0f) ? al : NEG_SLOPE * al;
    alpha[(size_t)e * HEADS + hd] = al;
    atomicMaxF(&amax[d * HEADS + hd], al);
  }
}

// ---------------- 6) ex = exp(alpha - amax[d]); segment sum -> denom --------
__global__ void exp_denom_kernel(const long long* __restrict__ ei,
    const float* __restrict__ amax, float* __restrict__ alpha,
    float* __restrict__ denom) {
  const int e = blockIdx.x * blockDim.x + threadIdx.x;
  if (e >= TOT_E) return;
  const long long d = (e < N_EDGES) ? ei[N_EDGES + e] : (long long)(e - N_EDGES);
#pragma unroll
  for (int hd = 0; hd < HEADS; ++hd) {
    const float ex = expf(alpha[(size_t)e * HEADS + hd] - amax[d * HEADS + hd]);
    alpha[(size_t)e * HEADS + hd] = ex;          // reuse buffer for ex
    atomicAdd(&denom[d * HEADS + hd], ex);
  }
}

// ---------------- 7) out[dst] += h[src] * att  (one wave per edge) ----------
__global__ void __launch_bounds__(256) scatter_kernel(const long long* __restrict__ ei,
    const float* __restrict__ h, const float* __restrict__ exbuf,
    const float* __restrict__ denom, float* __restrict__ out) {
  const int wave = blockIdx.x * 8 + (threadIdx.x >> 5);
  const int lane = threadIdx.x & 31;
  if (wave >= TOT_E) return;
  long long s, d;
  if (wave < N_EDGES) { s = ei[wave]; d = ei[N_EDGES + wave]; }
  else                { s = d = (long long)(wave - N_EDGES); }
  float att[HEADS];
#pragma unroll
  for (int hd = 0; hd < HEADS; ++hd)
    att[hd] = exbuf[(size_t)wave * HEADS + hd] / (denom[d * HEADS + hd] + 1e-16f);
  const float* hs = h + (size_t)s * HC;
  float* od = out + (size_t)d * HC;
  for (int c = lane; c < HC; c += 32)
    atomicAdd(&od[c], hs[c] * att[c >> 7]);      // c>>7 = head (CHAN=128)
}

// ---------------- 8) BatchNorm statistics ----------------
#define ROWS_PER_BLOCK 40
__global__ void bn_stats_kernel(const float* __restrict__ out,
                                float* __restrict__ chsum, float* __restrict__ chsq) {
  const int r0 = blockIdx.x * ROWS_PER_BLOCK;
  const int t = threadIdx.x;
  float s0 = 0.f, q0 = 0.f, s1 = 0.f, q1 = 0.f;
  for (int r = r0; r < r0 + ROWS_PER_BLOCK && r < N_NODES; ++r) {
    const float v0 = out[(size_t)r * HC + t];
    const float v1 = out[(size_t)r * HC + t + 256];
    s0 += v0; q0 += v0 * v0; s1 += v1; q1 += v1 * v1;
  }
  atomicAdd(&chsum[t], s0);        atomicAdd(&chsq[t], q0);
  atomicAdd(&chsum[t + 256], s1);  atomicAdd(&chsq[t + 256], q1);
}

__global__ void bn_finalize_kernel(const float* __restrict__ chsum,
                                   const float* __restrict__ chsq,
                                   float* __restrict__ mu, float* __restrict__ rstd) {
  const int c = blockIdx.x * blockDim.x + threadIdx.x;
  if (c >= HC) return;
  const float m = chsum[c] / (float)N_NODES;
  const float v = chsq[c] / (float)N_NODES - m * m;   // biased variance
  mu[c] = m;
  rstd[c] = rsqrtf(v + BN_EPS);
}

// Note: reference adds `bias` before BN; a constant per-channel shift is
// removed exactly by BN's mean subtraction, so it is skipped.
__global__ void bn_apply_kernel(float* __restrict__ out, const float* __restrict__ mu,
                                const float* __restrict__ rstd,
                                const float* __restrict__ gamma,
                                const float* __restrict__ beta) {
  const size_t total = (size_t)N_NODES * HC;
  const size_t stride = (size_t)gridDim.x * blockDim.x;
  for (size_t i = (size_t)blockIdx.x * blockDim.x + threadIdx.x; i < total; i += stride) {
    const int c = (int)(i & (HC - 1));
    out[i] = (out[i] - mu[c]) * rstd[c] * gamma[c] + beta[c];
  }
}

// ---------------- launcher ----------------
extern "C" void kernel_launch(void* const* d_in, const int* in_sizes, int n_in,
                              void* d_out, int out_size, void* d_ws, size_t ws_size,
                              hipStream_t stream) {
  const float*      x        = (const float*)d_in[0];
  const long long*  ei       = (const long long*)d_in[1];   // int64 [2,E]
  const float*      ea       = (const float*)d_in[2];
  const float*      W        = (const float*)d_in[3];
  const float*      W_edge   = (const float*)d_in[4];
  const float*      att_src  = (const float*)d_in[5];
  const float*      att_dst  = (const float*)d_in[6];
  const float*      att_edge = (const float*)d_in[7];
  // d_in[8] = bias (cancels under BN), d_in[9] = gamma, d_in[10] = beta
  const float*      gamma    = (const float*)d_in[9];
  const float*      beta     = (const float*)d_in[10];
  float*            out      = (float*)d_out;

  // workspace partition (floats); total ~47.7 MB
  float* ws      = (float*)d_ws;
  float* h       = ws;                                   // N*HC
  float* a_src   = h      + (size_t)N_NODES * HC;        // N*H
  float* a_dst   = a_src  + (size_t)N_NODES * HEADS;     // N*H
  float* alpha   = a_dst  + (size_t)N_NODES * HEADS;     // (E+N)*H
  float* amax    = alpha  + (size_t)TOT_E   * HEADS;     // N*H
  float* denom   = amax   + (size_t)N_NODES * HEADS;     // N*H
  float* easum   = denom  + (size_t)N_NODES * HEADS;     // EDIM
  float* M       = easum  + EDIM;                        // EDIM*HEADS
  float* loop_ae = M      + EDIM * HEADS;                // HEADS
  float* chsum   = loop_ae + HEADS;                      // HC
  float* chsq    = chsum  + HC;                          // HC
  float* mu      = chsq   + HC;                          // HC
  float* rstd    = mu     + HC;                          // HC

  init_kernel<<<4096, 256, 0, stream>>>(out, amax, denom, chsum, chsq, easum);

  // h = x @ W   (WMMA fp32): 1250 row-tiles x 4 col-groups = 5000 waves
  gemm_h_kernel<<<(N_NODES / 16 * 4 + 7) / 8, 256, 0, stream>>>(x, W, h);

  easum_kernel<<<512, 256, 0, stream>>>(ea, easum);
  setup_small_kernel<<<1, 256, 0, stream>>>(W_edge, att_edge, easum, M, loop_ae);

  srcdst_kernel<<<(N_NODES * HEADS + 7) / 8, 256, 0, stream>>>(h, att_src, att_dst,
                                                               a_src, a_dst);

  alpha_max_kernel<<<(TOT_E + 255) / 256, 256, 0, stream>>>(ei, ea, M, loop_ae,
                                                            a_src, a_dst, alpha, amax);
  exp_denom_kernel<<<(TOT_E + 255) / 256, 256, 0, stream>>>(ei, amax, alpha, denom);

  scatter_kernel<<<(TOT_E + 7) / 8, 256, 0, stream>>>(ei, h, alpha, denom, out);

  bn_stats_kernel<<<(N_NODES + ROWS_PER_BLOCK - 1) / ROWS_PER_BLOCK, 256, 0, stream>>>(
      out, chsum, chsq);
  bn_finalize_kernel<<<2, 256, 0, stream>>>(chsum, chsq, mu, rstd);
  bn_apply_kernel<<<4096, 256, 0, stream>>>(out, mu, rstd, gamma, beta);

  (void)in_sizes; (void)n_in; (void)out_size; (void)ws_size;
}